// OtsuThreshold_63127429317117
// MI455X (gfx1250) — compile-verified
//
#include <hip/hip_runtime.h>

// Otsu threshold, MI455X (gfx1250, wave32).
// Roofline: 256MB hist read + 256MB binarize read + 256MB write = 768MB
// @ 23.3 TB/s => ~33us. Bandwidth bound; the 256-bin scan work is done with
// V_WMMA_F32_16X16X4_F32 (the reference's tril/triu row-sums ARE triangular
// matmuls), in f32 to match jnp float32 cumsum rounding exactly.

typedef __attribute__((ext_vector_type(2))) float v2f;
typedef __attribute__((ext_vector_type(8))) float v8f;
typedef __attribute__((ext_vector_type(4))) int   v4i;

#define BINS 256
// d_ws layout: [0..255] u32 histogram, [256] i32 threshold
#define THRESH_OFF 256

// ---------------------------------------------------------------- zero ws ---
__global__ void otsu_zero_ws(unsigned int* __restrict__ ws) {
    int t = threadIdx.x;
    if (t < BINS + 1) ws[t] = 0u;
}

// -------------------------------------------------------------- histogram ---
// Per-wave LDS sub-histograms (8 waves/block) -> ds_add_u32 with low
// contention, then one global atomic per bin per block.
__global__ __launch_bounds__(256) void otsu_hist(const int* __restrict__ img,
                                                 long long n4,
                                                 unsigned int* __restrict__ gh) {
    __shared__ unsigned int sh[8][BINS];
    const int tid  = threadIdx.x;
    const int wave = tid >> 5;                 // wave32: 8 waves per 256 thr
    for (int i = tid; i < 8 * BINS; i += 256) ((unsigned int*)sh)[i] = 0u;
    __syncthreads();

    const v4i* img4 = (const v4i*)img;         // global_load_b128
    const long long stride = (long long)gridDim.x * 256;
    for (long long i = (long long)blockIdx.x * 256 + tid; i < n4; i += stride) {
        v4i p = img4[i];
        atomicAdd(&sh[wave][p[0] & 255], 1u);
        atomicAdd(&sh[wave][p[1] & 255], 1u);
        atomicAdd(&sh[wave][p[2] & 255], 1u);
        atomicAdd(&sh[wave][p[3] & 255], 1u);
    }
    __syncthreads();
    unsigned int sum = 0;
    #pragma unroll
    for (int w = 0; w < 8; ++w) sum += sh[w][tid];
    if (sum) atomicAdd(&gh[tid], sum);
}

// -------------------------------------------------- WMMA helper (f32, K=4) --
__device__ __forceinline__ v8f wmma4(v2f a, v2f b, v8f c) {
    // (neg_a, A, neg_b, B, c_mod, C, reuse_a, reuse_b)
    return __builtin_amdgcn_wmma_f32_16x16x4_f32(false, a, false, b,
                                                 (short)0, c, false, false);
}

// ------------------------------------------------------------- otsu stats ---
// Single wave (32 lanes, EXEC all-1s as WMMA requires).
// 256-vector x viewed as 16x16 matrix X[s][j] = x[16s+j].
//   Y = X * U   (U[j][c] = j<=c)       : within-segment inclusive scan
//   Z = W * T + Y (W[s][s']=s'<s, T[s'][*]=Y[s'][15]) : add segment offsets
// Layouts per ISA 7.12.2:
//   C/D f32 16x16 : lane holds (M = v + 8*half, N = lane&15) in slot v
//   A   f32 16x4  : lane holds (M = lane&15,  K = 2*half + v)
//   B   f32 4x16  : lane holds (K = v + 2*half, N = lane&15)
__global__ __launch_bounds__(32) void otsu_stats(const unsigned int* __restrict__ gh,
                                                 int* __restrict__ thresh_out,
                                                 float* __restrict__ out0) {
    __shared__ float sV[BINS];   // scan input
    __shared__ float sY[BINS];   // intermediate Y
    __shared__ float sCH[BINS];  // cumsum(hist)
    __shared__ float sCF[BINS];  // cumsum(i*hist)

    const int lane = threadIdx.x;
    const int half = lane >> 4;
    const int lo   = lane & 15;

    #pragma unroll
    for (int j = 0; j < 8; ++j) {
        int i = lane + 32 * j;
        sV[i] = (float)gh[i];
    }
    __syncthreads();

    float* dst[2] = { sCH, sCF };
    for (int pass = 0; pass < 2; ++pass) {
        if (pass == 1) {
            __syncthreads();
            #pragma unroll
            for (int j = 0; j < 8; ++j) {
                int i = lane + 32 * j;
                sV[i] = sV[i] * (float)i;      // fc = hist * levels
            }
            __syncthreads();
        }

        // ---- Y = X * U (4 WMMAs over K) ----
        v8f acc = {};
        #pragma unroll
        for (int k = 0; k < 4; ++k) {
            v2f a, b;
            a[0] = sV[16 * lo + 4 * k + 2 * half + 0];
            a[1] = sV[16 * lo + 4 * k + 2 * half + 1];
            b[0] = ((4 * k + 2 * half + 0) <= lo) ? 1.0f : 0.0f;
            b[1] = ((4 * k + 2 * half + 1) <= lo) ? 1.0f : 0.0f;
            acc = wmma4(a, b, acc);
        }
        __syncthreads();
        #pragma unroll
        for (int v = 0; v < 8; ++v) sY[16 * (v + 8 * half) + lo] = acc[v];
        __syncthreads();

        // ---- Z = W * T + Y (4 WMMAs over K) ----
        #pragma unroll
        for (int k = 0; k < 4; ++k) {
            v2f a, b;
            a[0] = ((4 * k + 2 * half + 0) < lo) ? 1.0f : 0.0f;
            a[1] = ((4 * k + 2 * half + 1) < lo) ? 1.0f : 0.0f;
            b[0] = sY[16 * (4 * k + 2 * half + 0) + 15];  // segment totals
            b[1] = sY[16 * (4 * k + 2 * half + 1) + 15];
            acc = wmma4(a, b, acc);
        }
        float* d = dst[pass];
        #pragma unroll
        for (int v = 0; v < 8; ++v) d[16 * (v + 8 * half) + lo] = acc[v];
        __syncthreads();
    }

    // ---- between-class variance + argmax (first-max, like jnp.argmax) ----
    const float totalN = sCH[255];
    const float totalF = sCF[255];
    float bestV = -1.0f;
    int   bestI = 0;
    #pragma unroll
    for (int j = 0; j < 8; ++j) {
        int   i  = lane + 32 * j;
        float nb = sCH[i], fb = sCF[i];
        float nw = totalN - nb, fw = totalF - fb;
        float mb = fb / nb;                    // 0/0 -> nan (torch semantics)
        float mw = fw / nw;
        float dd = mb - mw;
        float var = nb * nw * dd * dd;
        if (!(var == var)) var = 0.0f;         // nan -> 0
        if (var > bestV || (var == bestV && i < bestI)) { bestV = var; bestI = i; }
    }
    #pragma unroll
    for (int off = 16; off > 0; off >>= 1) {
        float ov = __shfl_down(bestV, off, 32);
        int   oi = __shfl_down(bestI, off, 32);
        if (ov > bestV || (ov == bestV && oi < bestI)) { bestV = ov; bestI = oi; }
    }
    if (lane == 0) {
        *thresh_out = bestI;
        *out0 = (float)bestI;                  // d_out[0] = thresh
    }
}

// --------------------------------------------------------------- binarize ---
// out starts at d_out+1 (4B aligned only), so b128 loads + b32 stores.
__global__ __launch_bounds__(256) void otsu_binarize(const int* __restrict__ img,
                                                     const int* __restrict__ thresh_p,
                                                     float* __restrict__ out,
                                                     long long n4) {
    const int th = *thresh_p;
    const v4i* img4 = (const v4i*)img;
    const long long stride = (long long)gridDim.x * 256;
    for (long long i = (long long)blockIdx.x * 256 + threadIdx.x; i < n4; i += stride) {
        v4i p = img4[i];
        long long b = i * 4;
        out[b + 0] = (p[0] <= th) ? 0.0f : 255.0f;
        out[b + 1] = (p[1] <= th) ? 0.0f : 255.0f;
        out[b + 2] = (p[2] <= th) ? 0.0f : 255.0f;
        out[b + 3] = (p[3] <= th) ? 0.0f : 255.0f;
    }
}

extern "C" void kernel_launch(void* const* d_in, const int* in_sizes, int n_in,
                              void* d_out, int out_size, void* d_ws, size_t ws_size,
                              hipStream_t stream) {
    const int* img = (const int*)d_in[0];
    const long long n  = (long long)in_sizes[0];   // 8192*8192
    const long long n4 = n >> 2;

    unsigned int* ws  = (unsigned int*)d_ws;
    float*        out = (float*)d_out;

    otsu_zero_ws<<<1, 288, 0, stream>>>(ws);
    otsu_hist<<<2048, 256, 0, stream>>>(img, n4, ws);
    otsu_stats<<<1, 32, 0, stream>>>(ws, (int*)(ws + THRESH_OFF), out);
    otsu_binarize<<<4096, 256, 0, stream>>>(img, (const int*)(ws + THRESH_OFF),
                                            out + 1, n4);
}